// TorchFeedForwardNetwork_82102594831011
// MI455X (gfx1250) — compile-verified
//
#include <hip/hip_runtime.h>
#include <math.h>

typedef __attribute__((ext_vector_type(2))) float v2f;
typedef __attribute__((ext_vector_type(8))) float v8f;

#define L_LAYERS 8
#define H_UNITS  512
#define K_FANIN  16
#define NIN      512
#define B_BATCH  8192
// ws layout: (4096, 8192) column-major "values": cols 0..511 = x^T,
// cols 512..4095 = outputs of layers 0..6. Layer 7 -> d_out (row-major).

__device__ __forceinline__ float fast_sigmoid(float z) {
  // 1/(1+e^-z): v_exp_f32 + v_rcp_f32, no IEEE div sequence
  return __builtin_amdgcn_rcpf(1.0f + __expf(-z));
}

// ---------------------------------------------------------------------------
// Tiled transpose: x (8192 x 512, row-major) -> wsT columns 0..511
// ---------------------------------------------------------------------------
__global__ __launch_bounds__(256) void transpose_x_kernel(
    const float* __restrict__ x, float* __restrict__ wsT)
{
  __shared__ float tile[32][33];
  const int tx = threadIdx.x, ty = threadIdx.y;   // (32, 8)
  const int c0 = blockIdx.x * 32;
  const int b0 = blockIdx.y * 32;

  #pragma unroll
  for (int i = 0; i < 32; i += 8)
    tile[ty + i][tx] = x[(size_t)(b0 + ty + i) * NIN + c0 + tx];
  __syncthreads();
  #pragma unroll
  for (int i = 0; i < 32; i += 8)
    wsT[(size_t)(c0 + ty + i) * B_BATCH + b0 + tx] = tile[tx][ty + i];
}

// ---------------------------------------------------------------------------
// One wave = one workgroup (wave32). Computes 16(h) x 128(batch) of a layer:
//   D[h,b] = sum_kk A'[h,kk] * B'[kk,b],  kk = h_local*16 + k
//   A' = block-diagonal weights (private regs), B' = column-major gathers.
// 4 batch-tiles per WMMA chunk share one address; chained f32 WMMA.
// ---------------------------------------------------------------------------
template <bool FINAL>
__global__ __launch_bounds__(32) void linknet_layer_kernel(
    const float* __restrict__ wsT,    // 4096 x 8192 column-major values
    const int*   __restrict__ idx_l,  // H x K indices (this layer)
    const float* __restrict__ w_l,    // H x K weights
    const float* __restrict__ b_l,    // H bias
    float* __restrict__ outp,         // FINAL ? row-major d_out : wsT base
    int out_col0)                     // first output column (non-final)
{
  __shared__ unsigned wOff[256];      // gather offsets: col * 8192 (elements)

  const int lane = threadIdx.x;       // 0..31
  const int hlf  = lane >> 4;
  const int l16  = lane & 15;

  const int ht = blockIdx.x & 31;     // h tile 0..31   (scalar)
  const int bg = blockIdx.x >> 5;     // 0..63          (scalar)
  const int hb = ht * 16;
  const int b0 = bg * 128;            // 8 batch tiles per wave

  for (int i = lane; i < 256; i += 32)
    wOff[i] = (unsigned)idx_l[hb * K_FANIN + i] * (unsigned)B_BATCH;
  __syncthreads();                    // single-wave group: S_NOP + dscnt wait

  // Lane's private block-diagonal weights: A'[M=l16,kk] nonzero iff kk>>4==l16;
  // lane only touches k with (k&3) in {2*hlf, 2*hlf+1}.
  float wpriv[8];
  #pragma unroll
  for (int j = 0; j < 8; ++j)
    wpriv[j] = w_l[(hb + l16) * K_FANIN + (j >> 1) * 4 + 2 * hlf + (j & 1)];

  // Bias for lane's D rows: M = r + 8*hlf
  float brow[8];
  #pragma unroll
  for (int r = 0; r < 8; ++r)
    brow[r] = b_l[hb + 8 * hlf + r];

  for (int g = 0; g < 2; ++g) {
    const float* __restrict__ pb = wsT + b0 + g * 64;   // scalar base (SGPR)
    v8f acc[4] = {};

    #pragma unroll 1
    for (int cb = 0; cb < 16; ++cb) {       // block-diagonal block = h_local
      #pragma unroll
      for (int cc = 0; cc < 4; ++cc) {      // 4 chunks of K=4 per block
        const int kk0 = cb * 16 + cc * 4 + 2 * hlf;
        const unsigned o0 = wOff[kk0]     + (unsigned)l16;
        const unsigned o1 = wOff[kk0 + 1] + (unsigned)l16;
        v2f a;                               // A': M = lane&15, K = 2*hlf+v
        a[0] = (l16 == cb) ? wpriv[cc * 2]     : 0.0f;
        a[1] = (l16 == cb) ? wpriv[cc * 2 + 1] : 0.0f;
        #pragma unroll
        for (int t = 0; t < 4; ++t) {        // 4 batch tiles, imm offsets
          v2f b;                             // B': N = lane&15, K = 2*hlf+v
          b[0] = pb[o0 + t * 16];
          b[1] = pb[o1 + t * 16];
          acc[t] = __builtin_amdgcn_wmma_f32_16x16x4_f32(
              false, a, false, b, (short)0, acc[t], false, false);
        }
      }
    }

    // D layout: VGPR r -> M(h_local) = r + 8*hlf, N(b_local) = l16
    #pragma unroll
    for (int t = 0; t < 4; ++t) {
      const int bb = b0 + g * 64 + t * 16;
      if (FINAL) {
        float* __restrict__ po = outp + (size_t)bb * H_UNITS + hb;  // scalar
        const unsigned vo = (unsigned)(l16 * H_UNITS + 8 * hlf);
        #pragma unroll
        for (int r = 0; r < 8; ++r)
          po[vo + (unsigned)r] = fast_sigmoid(acc[t][r] + brow[r]);
      } else {
        float* __restrict__ po = outp + (size_t)(out_col0 + hb) * B_BATCH + bb;
        const unsigned vo = (unsigned)(hlf * 8 * B_BATCH + l16);
        #pragma unroll
        for (int r = 0; r < 8; ++r)         // coalesced 64B per (r, half)
          po[vo + (unsigned)(r * B_BATCH)] = fast_sigmoid(acc[t][r] + brow[r]);
      }
    }
  }
}

extern "C" void kernel_launch(void* const* d_in, const int* in_sizes, int n_in,
                              void* d_out, int out_size, void* d_ws, size_t ws_size,
                              hipStream_t stream) {
  const float* x    = (const float*)d_in[0];  // (8192, 512)
  const int*   lidx = (const int*)  d_in[1];  // (8, 512, 16)
  const float* wgt  = (const float*)d_in[2];  // (8, 512, 16)
  const float* bias = (const float*)d_in[3];  // (8, 512)
  float* out = (float*)d_out;                 // (8192, 512)
  float* wsT = (float*)d_ws;                  // (4096, 8192) col-major values

  transpose_x_kernel<<<dim3(NIN / 32, B_BATCH / 32), dim3(32, 8), 0, stream>>>(x, wsT);

  for (int l = 0; l < L_LAYERS; ++l) {
    const int*   il = lidx + (size_t)l * H_UNITS * K_FANIN;
    const float* wl = wgt  + (size_t)l * H_UNITS * K_FANIN;
    const float* bl = bias + (size_t)l * H_UNITS;
    const int out_col0 = NIN + l * H_UNITS;
    if (l < L_LAYERS - 1)
      linknet_layer_kernel<false><<<dim3(2048), dim3(32), 0, stream>>>(
          wsT, il, wl, bl, wsT, out_col0);
    else
      linknet_layer_kernel<true><<<dim3(2048), dim3(32), 0, stream>>>(
          wsT, il, wl, bl, out, 0);
  }
}